// EnhancedSegmentationLoss_90134183674094
// MI455X (gfx1250) — compile-verified
//
#include <hip/hip_runtime.h>
#include <math.h>

#define B_      16
#define H_      1024
#define W_      1024
#define TS      32
#define HALO    34          // TS + 2 (1-px halo each side for 3x3 Sobel)
#define PITCH   35          // pad to dodge LDS bank conflicts
#define NUM_IDS 32
#define SMOOTHF 1e-6f

// workspace layout (floats):
//  [0]=focalSum [1]=sumP [2]=sumT [3]=sumPT [4]=magSum [5]=maskCnt [6]=cosMaskSum
#define WS_SUM  16
#define WS_CNT  (16 + B_*NUM_IDS)
#define WS_TOT  (16 + 2*B_*NUM_IDS)

typedef float v2f __attribute__((ext_vector_type(2)));
typedef float v8f __attribute__((ext_vector_type(8)));

__global__ void zero_ws_kernel(float* ws) {
  int i = blockIdx.x * blockDim.x + threadIdx.x;
  if (i < WS_TOT) ws[i] = 0.0f;
}

__device__ __forceinline__ float wave_sum32(float v) {
  #pragma unroll
  for (int off = 16; off > 0; off >>= 1) v += __shfl_xor(v, off, 32);
  return v;
}

__global__ void __launch_bounds__(256)
seg_loss_main(const float* __restrict__ pred,
              const float* __restrict__ tgt,
              const int*   __restrict__ ids,
              float*       __restrict__ ws) {
  __shared__ float s_p[HALO][PITCH];
  __shared__ float s_t[HALO][PITCH];
  __shared__ float histS[NUM_IDS];
  __shared__ float histC[NUM_IDS];
  __shared__ float part[8][16];

  const int b   = blockIdx.z;
  const int ty0 = blockIdx.y * TS;
  const int tx0 = blockIdx.x * TS;
  const int tid = threadIdx.x;
  const size_t base = (size_t)b * H_ * W_;

  if (tid < NUM_IDS) { histS[tid] = 0.0f; histC[tid] = 0.0f; }

  // --- load halo tile (replicate padding), p = clip(sigmoid(x)), t = clip(t) ---
  for (int i = tid; i < HALO * HALO; i += 256) {
    int ly = i / HALO, lx = i % HALO;
    int gy = ty0 + ly - 1; gy = gy < 0 ? 0 : (gy >= H_ ? H_ - 1 : gy);
    int gx = tx0 + lx - 1; gx = gx < 0 ? 0 : (gx >= W_ ? W_ - 1 : gx);
    size_t idx = base + (size_t)gy * W_ + gx;
    float x = pred[idx];
    float p = 1.0f / (1.0f + expf(-x));
    p = fminf(fmaxf(p, SMOOTHF), 1.0f - SMOOTHF);
    float t = fminf(fmaxf(tgt[idx], 0.0f), 1.0f);
    s_p[ly][lx] = p;
    s_t[ly][lx] = t;
  }
  __syncthreads();

  float aFocal = 0.0f, aP = 0.0f, aT = 0.0f, aPT = 0.0f;
  float aMag = 0.0f, aMask = 0.0f, aCos = 0.0f;

  const int lx  = tid & 31;
  const int ly0 = tid >> 5;   // 8 waves -> rows 0..7, strided by 8
  #pragma unroll
  for (int r = 0; r < 4; ++r) {
    const int ly = ly0 + r * 8;
    const float p = s_p[ly + 1][lx + 1];
    const float t = s_t[ly + 1][lx + 1];

    // focal (GAMMA=2, ALPHA=0.25); t is exactly 0 or 1 after clip of rounded input
    float bce = -(t * logf(p) + (1.0f - t) * log1pf(-p));
    float ptw = (t == 1.0f) ? p : (1.0f - p);
    float fw  = 1.0f - ptw; fw = fw * fw;
    float aw  = (t == 1.0f) ? 0.25f : 0.75f;
    aFocal += aw * fw * bce;
    aP += p; aT += t; aPT += p * t;

    // Sobel (cross-correlation, kx=[[-1,0,1],[-2,0,2],[-1,0,1]]/8, ky=kx^T)
    float tgx = ( s_t[ly][lx+2]   - s_t[ly][lx]
              + 2.0f*(s_t[ly+1][lx+2] - s_t[ly+1][lx])
              +   s_t[ly+2][lx+2] - s_t[ly+2][lx] ) * 0.125f;
    float tgy = ( s_t[ly+2][lx]   - s_t[ly][lx]
              + 2.0f*(s_t[ly+2][lx+1] - s_t[ly][lx+1])
              +   s_t[ly+2][lx+2] - s_t[ly][lx+2] ) * 0.125f;
    float pgx = ( s_p[ly][lx+2]   - s_p[ly][lx]
              + 2.0f*(s_p[ly+1][lx+2] - s_p[ly+1][lx])
              +   s_p[ly+2][lx+2] - s_p[ly+2][lx] ) * 0.125f;
    float pgy = ( s_p[ly+2][lx]   - s_p[ly][lx]
              + 2.0f*(s_p[ly+2][lx+1] - s_p[ly][lx+1])
              +   s_p[ly+2][lx+2] - s_p[ly][lx+2] ) * 0.125f;

    float tmag = sqrtf(tgx*tgx + tgy*tgy + 1e-6f);
    float pmag = sqrtf(pgx*pgx + pgy*pgy + 1e-6f);
    float bw   = 1.0f + 5.0f * tmag;
    float dm   = bw * (pmag - tmag);
    aMag += dm * dm;
    float mask = (tmag > 0.1f) ? 1.0f : 0.0f;
    aMask += mask;
    float cosv = (tgx * pgx + tgy * pgy) / (tmag * pmag);
    aCos += (1.0f - cosv) * mask;

    // per-(b,id) segment sums via LDS float atomics (ds_add_f32)
    int gid = ids[base + (size_t)(ty0 + ly) * W_ + (tx0 + lx)] & (NUM_IDS - 1);
    atomicAdd(&histS[gid], p);
    atomicAdd(&histC[gid], 1.0f);
  }

  // --- per-wave exact reduction of the 7 scalar accumulators ---
  aFocal = wave_sum32(aFocal); aP   = wave_sum32(aP);   aT   = wave_sum32(aT);
  aPT    = wave_sum32(aPT);    aMag = wave_sum32(aMag); aMask = wave_sum32(aMask);
  aCos   = wave_sum32(aCos);

  const int w = tid >> 5, lane = tid & 31;
  if (lane == 0) {
    part[w][0] = aFocal; part[w][1] = aP;   part[w][2] = aT;  part[w][3] = aPT;
    part[w][4] = aMag;   part[w][5] = aMask; part[w][6] = aCos;
    #pragma unroll
    for (int i = 7; i < 16; ++i) part[w][i] = 0.0f;
  }
  __syncthreads();

  // --- WMMA finisher (wave 0, EXEC all-ones): A = ones(16x4) so
  //     D[m][n] = sum_k B[k][n]; column n of B holds 4 wave-partials of
  //     quantity n (order along K irrelevant for a sum). Two chained WMMAs
  //     accumulate waves 0-3 then 4-7. Exact f32, RNE — same math as FMA tree.
  if (tid < 32) {
    const int n  = lane & 15;
    const int hi = lane >> 4;                // column n lives in lanes n and n+16
    v2f A;  A[0] = 1.0f;                A[1] = 1.0f;
    v2f B0; B0[0] = part[hi*2 + 0][n];  B0[1] = part[hi*2 + 1][n];
    v2f B1; B1[0] = part[hi*2 + 4][n];  B1[1] = part[hi*2 + 5][n];
    v8f C = {};
    C = __builtin_amdgcn_wmma_f32_16x16x4_f32(false, A, false, B0, (short)0, C, false, false);
    C = __builtin_amdgcn_wmma_f32_16x16x4_f32(false, A, false, B1, (short)0, C, false, false);
    // all D rows equal; lanes 0..15 of the first C register hold quantity n's total
    if (lane < 7) atomicAdd(&ws[lane], C[0]);
  }

  // --- per-block histogram flush to global (one atomic per bin per block) ---
  if (tid < NUM_IDS) {
    atomicAdd(&ws[WS_SUM + b * NUM_IDS + tid], histS[tid]);
    atomicAdd(&ws[WS_CNT + b * NUM_IDS + tid], histC[tid]);
  }
}

__global__ void seg_loss_finalize(const float* __restrict__ ws,
                                  float* __restrict__ out) {
  __shared__ float contrib[B_];
  const int tid = threadIdx.x;
  if (tid < B_) {
    const float* s = ws + WS_SUM + tid * NUM_IDS;
    const float* c = ws + WS_CNT + tid * NUM_IDS;
    float means[NUM_IDS];
    bool  valid[NUM_IDS];
    #pragma unroll
    for (int k = 0; k < NUM_IDS; ++k) {
      float cnt = c[k];
      means[k]  = s[k] / fmaxf(cnt, 1.0f);
      valid[k]  = (cnt > 0.0f) && (k > 0);
    }
    float np = 0.0f, cs = 0.0f;
    for (int i = 0; i < NUM_IDS; ++i) {
      if (!valid[i]) continue;
      for (int j = i + 1; j < NUM_IDS; ++j) {
        if (!valid[j]) continue;
        np += 1.0f;
        cs += expf(-fabsf(means[i] - means[j]));
      }
    }
    contrib[tid] = (np > 0.0f) ? cs / fmaxf(np, 1.0f) : 0.0f;
  }
  __syncthreads();
  if (tid == 0) {
    float csum = 0.0f;
    for (int b = 0; b < B_; ++b) csum += contrib[b];
    const float N     = (float)B_ * (float)H_ * (float)W_;
    float focal       = ws[0] / N;
    float dice        = 1.0f - (2.0f * ws[3] + SMOOTHF) / (ws[1] + ws[2] + SMOOTHF);
    float loss_mag    = ws[4] / N;
    float msum        = ws[5];
    float dir_loss    = (msum > 0.0f) ? (ws[6] / fmaxf(msum, 1.0f)) : 0.0f;
    float boundary    = loss_mag + dir_loss;
    float contrastive = csum / (float)B_;
    out[0] = focal + dice + 0.5f * boundary + 0.1f * contrastive;
  }
}

extern "C" void kernel_launch(void* const* d_in, const int* in_sizes, int n_in,
                              void* d_out, int out_size, void* d_ws, size_t ws_size,
                              hipStream_t stream) {
  const float* pred = (const float*)d_in[0];
  const float* tgt  = (const float*)d_in[1];
  const int*   ids  = (const int*)d_in[2];
  float* ws  = (float*)d_ws;
  float* out = (float*)d_out;

  zero_ws_kernel<<<(WS_TOT + 255) / 256, 256, 0, stream>>>(ws);
  dim3 grid(W_ / TS, H_ / TS, B_);
  seg_loss_main<<<grid, 256, 0, stream>>>(pred, tgt, ids, ws);
  seg_loss_finalize<<<1, 64, 0, stream>>>(ws, out);
}